// ViLLayer_53137335386840
// MI455X (gfx1250) — compile-verified
//
#include <hip/hip_runtime.h>
#include <math.h>
#include <stdint.h>

// ---- problem constants (match reference) ----
constexpr int CDIM  = 384;
constexpr int INNER = 768;
constexpr int NHM   = 4;      // mLSTM heads
constexpr int DHD   = 192;    // head dim
constexpr int NB    = 4;      // batch
constexpr int GH    = 32, GW = 32;
constexpr int SEQ   = 1024;   // GH*GW
constexpr float FEPS = 1e-6f;

typedef __attribute__((ext_vector_type(2))) float v2f;
typedef __attribute__((ext_vector_type(8))) float v8f;

// fp32 WMMA: D(16x16) = A(16x4) * B(4x16) + C  -- native fp32 matrix op on CDNA5
__device__ __forceinline__ v8f wmma4_f32(v2f a, v2f b, v8f c) {
  return __builtin_amdgcn_wmma_f32_16x16x4_f32(false, a, false, b, (short)0, c,
                                               false, false);
}

// ---- Tensor Data Mover support (guarded; falls back to cooperative loads) ----
#if __has_builtin(__builtin_amdgcn_tensor_load_to_lds) && \
    __has_builtin(__builtin_amdgcn_s_wait_tensorcnt)
#define USE_TDM 1
#else
#define USE_TDM 0
#endif

// K/V tile layout in LDS: 16 rows x 192 floats, with 4 pad DWORDs inserted
// every 64 DWORDs (TDM pad_interval=64, pad_amount=4) -> row stride 204 floats.
// 204 mod 64 = 12 => 16 row addresses land in 16 distinct banks.
constexpr int KVS   = 204;        // padded row stride (floats)
constexpr int TILEF = 16 * KVS;   // floats per K or V tile buffer
constexpr int SPP   = 17;         // padded P row stride (17*row mod 64 distinct)

#if USE_TDM
typedef unsigned int v4u __attribute__((ext_vector_type(4)));
typedef int v4i_t __attribute__((ext_vector_type(4)));
typedef int v8i_t __attribute__((ext_vector_type(8)));

// Issue one TDM descriptor: 2D tile, 16 rows x 192 fp32, row stride INNER,
// LDS padding 4 DWORDs per 64 DWORDs. Tracked with TENSORcnt.
__device__ __forceinline__ void tdm_load_kv_tile(const float* gsrc,
                                                 unsigned lds_off) {
  const unsigned long long ga = (unsigned long long)(uintptr_t)gsrc;
  v4u g0;
  g0.x = 1u;                                   // count=1 (user descriptor)
  g0.y = lds_off;                              // lds_addr (bytes)
  g0.z = (unsigned)ga;                         // global_addr[31:0]
  g0.w = (unsigned)((ga >> 32) & 0x01FFFFFFu)  // global_addr[56:32]
         | (2u << 30);                         // type = 2 ("image")
  v8i_t g1;
  g1[0] = (int)((2u << 16)     // data_size = 4 bytes
              | (1u << 20)     // pad_enable
              | (5u << 22)     // pad_interval code 5 = 64 DWORDs
              | (3u << 25));   // pad_amount  code 3 = 4 DWORDs
  g1[1] = (int)(192u << 16);   // tensor_dim0 = 192 (low 16 bits)
  g1[2] = 0;                   // tensor_dim0 hi = 0, tensor_dim1 lo = 0
  g1[3] = (int)(16u            // tensor_dim1 hi => dim1 = 1<<20 (no OOB)
              | (192u << 16)); // tile_dim0 = 192
  g1[4] = 16;                  // tile_dim1 = 16 rows; tile_dim2 = 0
  g1[5] = (int)768u;           // tensor_dim0_stride = INNER
  g1[6] = 0;
  g1[7] = 0;
  v4i_t gz4 = {0, 0, 0, 0};    // 2D tensor: groups 2/3 unused
  v8i_t gz8 = {0, 0, 0, 0, 0, 0, 0, 0};
  // 6-arg toolchain form: (g0, g1, g2, g3, extra, cpol)
  __builtin_amdgcn_tensor_load_to_lds(g0, g1, gz4, gz4, gz8, 0);
}
#endif

// =====================================================================
// 1/8. GEMM: C[M x N] = A[M x Ka] @ W[N x Ka]^T + bias, column-split output.
//     4 waves/block, each wave owns one 16x16 tile; K-loop of 4 (fp32 WMMA).
// =====================================================================
__global__ void __launch_bounds__(128)
gemm_split_kernel(const float* __restrict__ A, const float* __restrict__ Wt,
                  const float* __restrict__ bias, float* __restrict__ C0,
                  float* __restrict__ C1, int M, int N, int Ka, int Nsplit) {
  const int lane = threadIdx.x & 31;
  const int wv   = threadIdx.x >> 5;
  const int hl   = lane >> 4;   // half-wave select (K offset 0 / 2)
  const int mr   = lane & 15;   // row (A) / col (B,C)
  const int m0   = blockIdx.y * 16;
  const int n0   = blockIdx.x * 64 + wv * 16;
  const float* Arow = A  + (size_t)(m0 + mr) * Ka;
  const float* Wrow = Wt + (size_t)(n0 + mr) * Ka;
  v8f acc = {};
  for (int kb = 0; kb < Ka; kb += 4) {
    const int k0 = kb + 2 * hl;
    v2f a; a.x = Arow[k0]; a.y = Arow[k0 + 1];
    v2f b; b.x = Wrow[k0]; b.y = Wrow[k0 + 1];
    acc = wmma4_f32(a, b, acc);
  }
  const int n = n0 + mr;
  const float bv = bias[n];
#pragma unroll
  for (int r = 0; r < 8; ++r) {
    const int m = m0 + r + 8 * hl;
    const float val = acc[r] + bv;
    if (n < Nsplit) C0[(size_t)m * Nsplit + n] = val;
    else            C1[(size_t)m * (N - Nsplit) + (n - Nsplit)] = val;
  }
}

// =====================================================================
// 2. depthwise 3x3 conv (SAME) + bias + SiLU
// =====================================================================
__global__ void __launch_bounds__(256)
conv_silu_kernel(const float* __restrict__ xm, const float* __restrict__ cw,
                 const float* __restrict__ cb, float* __restrict__ ca) {
  const int idx = blockIdx.x * 256 + threadIdx.x;
  const int c = idx % INNER;
  const int t = idx / INNER;
  const int s = t % SEQ;
  const int b = t / SEQ;
  const int y = s >> 5, x = s & 31;
  float acc = 0.f;
#pragma unroll
  for (int ky = 0; ky < 3; ++ky) {
    const int yy = y + ky - 1;
    if (yy < 0 || yy >= GH) continue;
#pragma unroll
    for (int kx = 0; kx < 3; ++kx) {
      const int xx = x + kx - 1;
      if (xx < 0 || xx >= GW) continue;
      acc += xm[((size_t)b * SEQ + yy * GW + xx) * INNER + c] *
             cw[(ky * 3 + kx) * INNER + c];
    }
  }
  acc += cb[c];
  ca[idx] = acc / (1.f + __expf(-acc));   // SiLU
}

// =====================================================================
// 3. headwise block-diagonal 4x4 projections: q,k from conv_act; v from x_m
// =====================================================================
__global__ void __launch_bounds__(256)
qkv_kernel(const float* __restrict__ ca, const float* __restrict__ xm,
           const float* __restrict__ Wq, const float* __restrict__ bq,
           const float* __restrict__ Wk, const float* __restrict__ bk,
           const float* __restrict__ Wv, const float* __restrict__ bv,
           float* __restrict__ q, float* __restrict__ k, float* __restrict__ v) {
  const int idx = blockIdx.x * 256 + threadIdx.x;
  const int c = idx % INNER;
  const size_t base = (size_t)(idx - c);
  const int hq = c >> 2, o = c & 3;
  const float* cav = ca + base + (hq << 2);
  const float* xmv = xm + base + (hq << 2);
  float aq = 0.f, ak = 0.f, av = 0.f;
#pragma unroll
  for (int d = 0; d < 4; ++d) {
    const int wi = (hq * 4 + o) * 4 + d;
    const float cx = cav[d], mx = xmv[d];
    aq += cx * Wq[wi];
    ak += cx * Wk[wi];
    av += mx * Wv[wi];
  }
  q[idx] = aq + bq[c];
  k[idx] = ak + bk[c];
  v[idx] = av + bv[c];
}

// =====================================================================
// 4. gates: ig/fg[b,h,s] = [q|k|v] . W_{ig,fg}[h] + bias   (8 dots of 2304)
// =====================================================================
__global__ void __launch_bounds__(256)
gates_kernel(const float* __restrict__ q, const float* __restrict__ k,
             const float* __restrict__ v, const float* __restrict__ Wig,
             const float* __restrict__ big, const float* __restrict__ Wfg,
             const float* __restrict__ bfg, float* __restrict__ ig,
             float* __restrict__ fg) {
  __shared__ float red[8][256];
  const int t = threadIdx.x;
  const int bs = blockIdx.x;             // b*SEQ + s
  const size_t base = (size_t)bs * INNER;
  float p[8] = {0.f, 0.f, 0.f, 0.f, 0.f, 0.f, 0.f, 0.f};
  for (int c = t; c < 3 * INNER; c += 256) {
    const float val = (c < INNER)       ? q[base + c]
                    : (c < 2 * INNER)   ? k[base + c - INNER]
                                        : v[base + c - 2 * INNER];
#pragma unroll
    for (int h = 0; h < NHM; ++h) {
      p[h]     += val * Wig[h * 3 * INNER + c];
      p[4 + h] += val * Wfg[h * 3 * INNER + c];
    }
  }
#pragma unroll
  for (int o = 0; o < 8; ++o) red[o][t] = p[o];
  __syncthreads();
  for (int off = 128; off > 0; off >>= 1) {
    if (t < off)
#pragma unroll
      for (int o = 0; o < 8; ++o) red[o][t] += red[o][t + off];
    __syncthreads();
  }
  if (t < NHM) {
    const int b = bs / SEQ, s = bs % SEQ;
    ig[((size_t)(b * NHM + t)) * SEQ + s] = red[t][0] + big[t];
    fg[((size_t)(b * NHM + t)) * SEQ + s] = red[4 + t][0] + bfg[t];
  }
}

// =====================================================================
// 5. per-(b,h) scans: lfc = cumsum(log_sigmoid(fg)); g = ig - lfc;
//    M = inclusive prefix-max of g.  m_i = lfc[i] + M[i] -> no online rescale.
// =====================================================================
__global__ void __launch_bounds__(1024)
scan_kernel(const float* __restrict__ ig, const float* __restrict__ fg,
            float* __restrict__ lfc, float* __restrict__ g,
            float* __restrict__ Mx) {
  __shared__ float sh[SEQ];
  const int t = threadIdx.x;
  const size_t base = (size_t)blockIdx.x * SEQ;
  const float f = fg[base + t];
  const float ls = (f > 0.f) ? -log1pf(__expf(-f)) : f - log1pf(__expf(f));
  sh[t] = ls;
  __syncthreads();
  for (int off = 1; off < SEQ; off <<= 1) {      // Hillis-Steele inclusive sum
    const float o = (t >= off) ? sh[t - off] : 0.f;
    __syncthreads();
    sh[t] += o;
    __syncthreads();
  }
  const float l = sh[t];
  lfc[base + t] = l;
  const float gv = ig[base + t] - l;
  g[base + t] = gv;
  __syncthreads();
  sh[t] = gv;
  __syncthreads();
  for (int off = 1; off < SEQ; off <<= 1) {      // inclusive prefix-max
    const float o = (t >= off) ? sh[t - off] : -1e30f;
    __syncthreads();
    sh[t] = fmaxf(sh[t], o);
    __syncthreads();
  }
  Mx[base + t] = sh[t];
}

// =====================================================================
// 6. mLSTM flash kernel. 4 waves/block, 64 rows/block, wave owns 16 rows.
//    Double-buffered K/V tiles prefetched by the Tensor Data Mover while
//    the block runs QK^T (48 wmma) -> decay/mask -> P@V (48+4 wmma).
// =====================================================================
__global__ void __launch_bounds__(128)
mlstm_attn_kernel(const float* __restrict__ q, const float* __restrict__ k,
                  const float* __restrict__ v, const float* __restrict__ lfc,
                  const float* __restrict__ g, const float* __restrict__ Mx,
                  float* __restrict__ hraw) {
  __shared__ float sKt[2][TILEF];
  __shared__ float sVt[2][TILEF];
  __shared__ float sP[4][16 * SPP];
  __shared__ float sLfc[64], sM[64];

  const int lane = threadIdx.x & 31;
  const int wv   = threadIdx.x >> 5;
  const int hl   = lane >> 4;
  const int mr   = lane & 15;
  const int bh   = blockIdx.y;          // b*NHM + h
  const int b    = bh >> 2;
  const int h    = bh & 3;
  const int rowbase = blockIdx.x * 64;
  const int i0      = rowbase + wv * 16;
  const size_t scl  = (size_t)bh * SEQ;
  const float inv_sqrt = 0.0721687836487032f;   // 1/sqrt(192)

  if (threadIdx.x < 64) {
    sLfc[threadIdx.x] = lfc[scl + rowbase + threadIdx.x];
    sM[threadIdx.x]   = Mx[scl + rowbase + threadIdx.x];
  }

  v8f zero = {};
  v8f accd[12];
#pragma unroll
  for (int d = 0; d < 12; ++d) accd[d] = zero;
  v8f accS = zero;
  v2f ones; ones.x = 1.f; ones.y = 1.f;

  const size_t hbase = (size_t)b * SEQ * INNER + (size_t)h * DHD;
  const int lastTile   = (rowbase + 63) >> 4;
  const int myLastTile = (i0 + 15) >> 4;

  auto issue_tile = [&](int t, int bsel) {
    const float* kt = k + hbase + (size_t)(t * 16) * INNER;
    const float* vt = v + hbase + (size_t)(t * 16) * INNER;
#if USE_TDM
    if (wv == 0) {   // one wave drives the TDM; EXEC is ignored by tensor ops
      tdm_load_kv_tile(kt, (unsigned)(uintptr_t)&sKt[bsel][0]);
      tdm_load_kv_tile(vt, (unsigned)(uintptr_t)&sVt[bsel][0]);
    }
#else
    for (int e = threadIdx.x; e < 16 * DHD; e += 128) {
      const int jr = e / DHD, d = e % DHD;
      const int lo = jr * KVS + d + 4 * (d >> 6);   // TDM-equivalent padding
      sKt[bsel][lo] = kt[(size_t)jr * INNER + d];
      sVt[bsel][lo] = vt[(size_t)jr * INNER + d];
    }
#endif
  };

  issue_tile(0, 0);                       // prologue prefetch
  for (int t = 0; t <= lastTile; ++t) {
    const int cur = t & 1;
    const bool haveNext = t < lastTile;
    if (haveNext) issue_tile(t + 1, cur ^ 1);   // prefetch next tile
#if USE_TDM
    if (wv == 0) {  // in-order TENSORcnt: <=2 outstanding => tile t complete
      if (haveNext) __builtin_amdgcn_s_wait_tensorcnt(2);
      else          __builtin_amdgcn_s_wait_tensorcnt(0);
    }
#endif
    __syncthreads();                      // tile t visible to all waves

    if (t <= myLastTile) {                // wave-uniform causal skip
      const int j0 = t * 16;
      const float* Kc = &sKt[cur][0];
      const float* Vc = &sVt[cur][0];
      // ---- QK^T : 16x16 tile over DH=192 ----
      v8f pacc = zero;
      const float* qrow = q + hbase + (size_t)(i0 + mr) * INNER;
      for (int kb = 0; kb < DHD; kb += 4) {
        const int k0 = kb + 2 * hl;
        const int ko = mr * KVS + k0 + 4 * (k0 >> 6);
        v2f a;  a.x  = qrow[k0];  a.y  = qrow[k0 + 1];
        v2f bb; bb.x = Kc[ko];    bb.y = Kc[ko + 1];
        pacc = wmma4_f32(a, bb, pacc);
      }
      // ---- decay weight + causal mask, in fragment layout ----
      const float gj = g[scl + j0 + mr];
#pragma unroll
      for (int r = 0; r < 8; ++r) {
        const int row = r + 8 * hl;
        const int i = i0 + row, j = j0 + mr;
        const float w =
            (j <= i) ? __expf(gj - sM[i - rowbase]) * inv_sqrt : 0.f;
        sP[wv][row * SPP + mr] = pacc[r] * w;
      }
      // ---- P @ V (12 d-tiles) + row-sum tile via all-ones B ----
#pragma unroll
      for (int kb = 0; kb < 16; kb += 4) {
        const int k0 = kb + 2 * hl;
        v2f a; a.x = sP[wv][mr * SPP + k0]; a.y = sP[wv][mr * SPP + k0 + 1];
#pragma unroll
        for (int dt = 0; dt < 12; ++dt) {
          const int d = dt * 16 + mr;
          const int dp = d + 4 * (d >> 6);
          v2f bb;
          bb.x = Vc[k0 * KVS + dp];
          bb.y = Vc[(k0 + 1) * KVS + dp];
          accd[dt] = wmma4_f32(a, bb, accd[dt]);
        }
        accS = wmma4_f32(a, ones, accS);
      }
    }
    __syncthreads();                      // all done reading buffer `cur`
  }

  // ---- epilogue: normalizer = max(|sum|, exp(-lfc - M)); divide; store ----
  const size_t obase = (size_t)bh * SEQ * DHD;
#pragma unroll
  for (int r = 0; r < 8; ++r) {
    const int row = r + 8 * hl;
    const int i = i0 + row;
    const float s = accS[r];
    const float nrm =
        fmaxf(fabsf(s), __expf(-sLfc[i - rowbase] - sM[i - rowbase]));
    const float scale = 1.f / (nrm + FEPS);
#pragma unroll
    for (int dt = 0; dt < 12; ++dt)
      hraw[obase + (size_t)i * DHD + dt * 16 + mr] = accd[dt][r] * scale;
  }
}

// =====================================================================
// 7. multi-head LayerNorm (over DH) + skip*conv_act + silu(z) gating
// =====================================================================
__global__ void __launch_bounds__(192)
norm_gate_kernel(const float* __restrict__ hraw, const float* __restrict__ nw,
                 const float* __restrict__ nb, const float* __restrict__ skip,
                 const float* __restrict__ ca, const float* __restrict__ z,
                 float* __restrict__ hstate) {
  __shared__ float s1[256], s2[256];
  const int t = threadIdx.x;                 // d in [0,192)
  const int id = blockIdx.x;                 // (b*SEQ + s)*NHM + h
  const int h = id & 3;
  const int s = (id >> 2) & (SEQ - 1);
  const int b = id >> 12;
  const float hv = hraw[(((size_t)(b * NHM + h)) * SEQ + s) * DHD + t];
  s1[t] = hv; s2[t] = hv * hv;
  if (t < 64) { s1[192 + t] = 0.f; s2[192 + t] = 0.f; }
  __syncthreads();
  for (int off = 128; off > 0; off >>= 1) {
    if (t < off) { s1[t] += s1[t + off]; s2[t] += s2[t + off]; }
    __syncthreads();
  }
  const float mean = s1[0] * (1.f / DHD);
  const float var  = s2[0] * (1.f / DHD) - mean * mean;
  const int c = h * DHD + t;
  const float hn = (hv - mean) * rsqrtf(var + 1e-5f) * nw[c] + nb[c];
  const size_t idx = ((size_t)b * SEQ + s) * INNER + c;
  const float hs = hn + skip[c] * ca[idx];
  const float zv = z[idx];
  hstate[idx] = hs * (zv / (1.f + __expf(-zv)));
}

// =====================================================================
extern "C" void kernel_launch(void* const* d_in, const int* in_sizes, int n_in,
                              void* d_out, int out_size, void* d_ws,
                              size_t ws_size, hipStream_t stream) {
  const float* x    = (const float*)d_in[0];
  const float* W_up = (const float*)d_in[1];
  const float* b_up = (const float*)d_in[2];
  const float* Wq   = (const float*)d_in[3];
  const float* bq   = (const float*)d_in[4];
  const float* Wk   = (const float*)d_in[5];
  const float* bk   = (const float*)d_in[6];
  const float* Wv   = (const float*)d_in[7];
  const float* bv   = (const float*)d_in[8];
  const float* cw   = (const float*)d_in[9];
  const float* cb   = (const float*)d_in[10];
  const float* Wig  = (const float*)d_in[11];
  const float* big  = (const float*)d_in[12];
  const float* Wfg  = (const float*)d_in[13];
  const float* bfg  = (const float*)d_in[14];
  const float* nw   = (const float*)d_in[15];
  const float* nbq  = (const float*)d_in[16];
  const float* skip = (const float*)d_in[17];
  const float* W_dn = (const float*)d_in[18];
  const float* b_dn = (const float*)d_in[19];
  float* out = (float*)d_out;
  (void)in_sizes; (void)n_in; (void)out_size; (void)ws_size;

  char* base = (char*)d_ws;
  size_t off = 0;
  auto carve = [&](size_t nfloats) -> float* {
    float* p = (float*)(base + off);
    off += ((nfloats * sizeof(float)) + 255) & ~(size_t)255;
    return p;
  };
  const size_t NTOK = (size_t)NB * SEQ;               // 4096 tokens
  float* xm     = carve(NTOK * INNER);
  float* zbuf   = carve(NTOK * INNER);
  float* ca     = carve(NTOK * INNER);
  float* qb     = carve(NTOK * INNER);
  float* kb     = carve(NTOK * INNER);
  float* vb     = carve(NTOK * INNER);
  float* igb    = carve((size_t)NB * NHM * SEQ);
  float* fgb    = carve((size_t)NB * NHM * SEQ);
  float* lfcb   = carve((size_t)NB * NHM * SEQ);
  float* gbuf   = carve((size_t)NB * NHM * SEQ);
  float* Mxb    = carve((size_t)NB * NHM * SEQ);
  float* hraw   = carve((size_t)NB * NHM * SEQ * DHD);
  float* hstate = carve(NTOK * INNER);

  // 1. up-projection (split into x_m | z)
  gemm_split_kernel<<<dim3(2 * INNER / 64, NTOK / 16), 128, 0, stream>>>(
      x, W_up, b_up, xm, zbuf, (int)NTOK, 2 * INNER, CDIM, INNER);
  // 2. depthwise conv + SiLU
  conv_silu_kernel<<<(int)(NTOK * INNER / 256), 256, 0, stream>>>(xm, cw, cb, ca);
  // 3. headwise q/k/v
  qkv_kernel<<<(int)(NTOK * INNER / 256), 256, 0, stream>>>(
      ca, xm, Wq, bq, Wk, bk, Wv, bv, qb, kb, vb);
  // 4. gates
  gates_kernel<<<(int)NTOK, 256, 0, stream>>>(qb, kb, vb, Wig, big, Wfg, bfg,
                                              igb, fgb);
  // 5. scans (lfc, g, prefix-max M)
  scan_kernel<<<NB * NHM, SEQ, 0, stream>>>(igb, fgb, lfcb, gbuf, Mxb);
  // 6. mLSTM flash attention (TDM double-buffered K/V tiles)
  mlstm_attn_kernel<<<dim3(SEQ / 64, NB * NHM), 128, 0, stream>>>(
      qb, kb, vb, lfcb, gbuf, Mxb, hraw);
  // 7. head-LN + skip + output gating
  norm_gate_kernel<<<(int)(NTOK * NHM), DHD, 0, stream>>>(
      hraw, nw, nbq, skip, ca, zbuf, hstate);
  // 8. down-projection
  gemm_split_kernel<<<dim3(CDIM / 64, NTOK / 16), 128, 0, stream>>>(
      hstate, W_dn, b_dn, out, out, (int)NTOK, CDIM, INNER, CDIM);
}